// SheafAttention_31842887533241
// MI455X (gfx1250) — compile-verified
//
#include <hip/hip_runtime.h>
#include <hip/hip_bf16.h>

typedef __attribute__((ext_vector_type(2))) float v2f;
typedef __attribute__((ext_vector_type(8))) float v8f;

#define TS_LOG2 20
#define TS (1u << TS_LOG2)
#define EMPTY_KEY 0xFFFFFFFFu

__device__ __forceinline__ unsigned hash_slot(unsigned k) {
    return (k * 2654435761u) >> (32 - TS_LOG2);
}

// ---- K0: init hash table + degree (identity diag contributes exactly 1.0/row)
__global__ void k_init(unsigned* __restrict__ keys, int* __restrict__ vals,
                       float* __restrict__ deg, int nd) {
    int i = blockIdx.x * blockDim.x + threadIdx.x;
    if (i < (int)TS) { keys[i] = EMPTY_KEY; vals[i] = 0x7FFFFFFF; }
    if (i < nd) deg[i] = 1.0f;
}

// ---- K1: insert forward-edge hashes; atomicMin == stable-sort first-dup semantics
__global__ void k_insert(const int* __restrict__ row, const int* __restrict__ col,
                         int E, int n, unsigned* __restrict__ keys, int* __restrict__ vals) {
    int e = blockIdx.x * blockDim.x + threadIdx.x;
    if (e >= E) return;
    unsigned h = (unsigned)(row[e] * n + col[e]);
    unsigned slot = hash_slot(h);
    for (;;) {
        unsigned prev = atomicCAS(&keys[slot], EMPTY_KEY, h);
        if (prev == EMPTY_KEY || prev == h) { atomicMin(&vals[slot], e); break; }
        slot = (slot + 1u) & (TS - 1u);
    }
}

// ---- K2: reverse-edge lookup (self-fallback when absent, as in reference)
__global__ void k_lookup(const int* __restrict__ row, const int* __restrict__ col,
                         int E, int n, const unsigned* __restrict__ keys,
                         const int* __restrict__ vals, int* __restrict__ rev) {
    int e = blockIdx.x * blockDim.x + threadIdx.x;
    if (e >= E) return;
    unsigned h = (unsigned)(col[e] * n + row[e]);
    unsigned slot = hash_slot(h);
    int r = e;
    for (;;) {
        unsigned k = keys[slot];
        if (k == h) { r = vals[slot]; break; }
        if (k == EMPTY_KEY) break;
        slot = (slot + 1u) & (TS - 1u);
    }
    rev[e] = r;
}

// ---- K3: per-wave, 2 edges packed block-diagonally into one 16x16 f32 WMMA tile.
// P = A^T * B with A = F_e (8x8), B = F_rev (8x8); K=8 done as two chained 16x16x4.
// A-operand 16x4 layout: lane m=L&15 -> M=m, lane-half selects K pair {0,1}/{2,3}.
// B-operand 4x16 layout mirrors it; since tile column n == m per lane and
// i = n&7 == m&7, A and B loads share identical per-lane offsets (different base).
__global__ __launch_bounds__(128)
void k_edge_blocks(const float* __restrict__ rm, const int* __restrict__ row,
                   const int* __restrict__ rev, float* __restrict__ out,
                   float* __restrict__ deg, int E) {
    const int lane = threadIdx.x & 31;
    const long wid = (long)blockIdx.x * (blockDim.x >> 5) + (threadIdx.x >> 5);
    const long e0 = 2 * wid;
    const long e1 = e0 + 1;
    if (e0 >= (long)E) return;                 // wave-uniform: EXEC stays all-1s
    const bool haveE1 = (e1 < (long)E);

    const int m    = lane & 15;
    const int half = lane >> 4;                // selects K pair within a step
    const int i    = m & 7;                    // matrix row/col within 8x8 block
    const long eA  = (m < 8) ? e0 : (haveE1 ? e1 : e0);
    const long rA  = eA * 64;
    const long rB  = (long)rev[eA] * 64;

    v8f c = {};
#pragma unroll
    for (int s = 0; s < 2; ++s) {
        const int j0 = 4 * s + 2 * half;       // contraction index j (two per lane)
        v2f a, b;
        a[0] = rm[rA + (long)(j0 + 0) * 8 + i];   // A^T: A_mat[m, k] = F[j, i]
        a[1] = rm[rA + (long)(j0 + 1) * 8 + i];
        b[0] = rm[rB + (long)(j0 + 0) * 8 + i];   // B_mat[k, n] = Frev[j, n&7]
        b[1] = rm[rB + (long)(j0 + 1) * 8 + i];
        c = __builtin_amdgcn_wmma_f32_16x16x4_f32(false, a, false, b,
                                                  (short)0, c, false, false);
    }

    // |P| row sums: butterfly over the 8-lane column groups {0..7} and {24..31}
    v8f s;
#pragma unroll
    for (int r = 0; r < 8; ++r) s[r] = fabsf(c[r]);
#pragma unroll
    for (int mask = 1; mask < 8; mask <<= 1) {
#pragma unroll
        for (int r = 0; r < 8; ++r) s[r] += __shfl_xor(s[r], mask, 32);
    }

    // D layout: VGPR r, lanes 0-15 -> D[r, n]; lanes 16-31 -> D[r+8, n].
    // Edge0 block = lanes 0..7 (n<8); edge1 block = lanes 24..31 (n>=8).
    if (half == 0 && m < 8) {
        const long base = e0 * 64 + m;         // out[e0, r, n] row-major
#pragma unroll
        for (int r = 0; r < 8; ++r) out[base + (long)r * 8] = c[r];
    }
    if (half == 1 && m >= 8 && haveE1) {
        const long base = e1 * 64 + (m - 8);
#pragma unroll
        for (int r = 0; r < 8; ++r) out[base + (long)r * 8] = c[r];
    }
    if (lane == 0) {
        const long db = (long)row[e0] * 8;
#pragma unroll
        for (int r = 0; r < 8; ++r) atomicAdd(&deg[db + r], s[r]);
    }
    if (lane == 24 && haveE1) {
        const long db = (long)row[e1] * 8;
#pragma unroll
        for (int r = 0; r < 8; ++r) atomicAdd(&deg[db + r], s[r]);
    }
}

// ---- K4: dinv = deg^-1/2 (deg >= 1 always, keep reference's guard anyway)
__global__ void k_dinv(const float* __restrict__ deg, float* __restrict__ dinv, int nd) {
    int i = blockIdx.x * blockDim.x + threadIdx.x;
    if (i >= nd) return;
    float d = deg[i];
    dinv[i] = (d > 0.0f) ? rsqrtf(d) : 0.0f;
}

// ---- K5: in-place symmetric normalization of off-diagonal blocks
__global__ void k_norm_off(float* __restrict__ out, const int* __restrict__ row,
                           const int* __restrict__ col, const float* __restrict__ dinv,
                           long total) {
    long idx = (long)blockIdx.x * blockDim.x + threadIdx.x;
    if (idx >= total) return;
    long e = idx >> 6;
    int t = (int)(idx & 63);
    int i = t >> 3, j = t & 7;
    out[idx] *= dinv[(long)row[e] * 8 + i] * dinv[(long)col[e] * 8 + j];
}

// ---- K6: normalized identity diagonal blocks
__global__ void k_diag(float* __restrict__ out, const float* __restrict__ dinv,
                       long baseOff, long total) {
    long idx = (long)blockIdx.x * blockDim.x + threadIdx.x;
    if (idx >= total) return;
    long v = idx >> 6;
    int t = (int)(idx & 63);
    int i = t >> 3, j = t & 7;
    float di = dinv[v * 8 + i];
    out[baseOff + idx] = (i == j) ? di * di : 0.0f;
}

extern "C" void kernel_launch(void* const* d_in, const int* in_sizes, int n_in,
                              void* d_out, int out_size, void* d_ws, size_t ws_size,
                              hipStream_t stream) {
    const float* rm  = (const float*)d_in[0];      // (E, 8, 8) f32
    const int*   ei  = (const int*)d_in[1];        // (2, E) i32
    float*       out = (float*)d_out;              // (E+n)*64 f32

    const int E  = in_sizes[1] / 2;
    const int n  = out_size / 64 - E;              // N_NODES
    const int nd = n * 8;
    const int* row = ei;
    const int* col = ei + E;

    // workspace carve-up (256B aligned)
    char* ws = (char*)d_ws;
    size_t off = 0;
    auto take = [&](size_t bytes) { char* p = ws + off; off = (off + bytes + 255) & ~(size_t)255; return p; };
    unsigned* keys = (unsigned*)take(sizeof(unsigned) * TS);
    int*      vals = (int*)take(sizeof(int) * TS);
    int*      rev  = (int*)take(sizeof(int) * (size_t)E);
    float*    deg  = (float*)take(sizeof(float) * (size_t)nd);
    float*    dinv = (float*)take(sizeof(float) * (size_t)nd);
    (void)ws_size; (void)n_in;

    const int B = 256;
    int initN = (int)TS > nd ? (int)TS : nd;
    k_init<<<(initN + B - 1) / B, B, 0, stream>>>(keys, vals, deg, nd);
    k_insert<<<(E + B - 1) / B, B, 0, stream>>>(row, col, E, n, keys, vals);
    k_lookup<<<(E + B - 1) / B, B, 0, stream>>>(row, col, E, n, keys, vals, rev);

    // 2 edges per wave, 4 waves per 128-thread block
    long nWaves  = ((long)E + 1) / 2;
    long nBlocks = (nWaves + 3) / 4;
    k_edge_blocks<<<(int)nBlocks, 128, 0, stream>>>(rm, row, rev, out, deg, E);

    k_dinv<<<(nd + B - 1) / B, B, 0, stream>>>(deg, dinv, nd);

    long offTotal = (long)E * 64;
    k_norm_off<<<(int)((offTotal + B - 1) / B), B, 0, stream>>>(out, row, col, dinv, offTotal);

    long diagTotal = (long)n * 64;
    k_diag<<<(int)((diagTotal + B - 1) / B), B, 0, stream>>>(out, dinv, offTotal, diagTotal);
}